// AI4Urban_29334626632294
// MI455X (gfx1250) — compile-verified
//
#include <hip/hip_runtime.h>
#include <stdint.h>

// ---------------------------------------------------------------------------
// AI4Urban forward step on MI455X (gfx1250, wave32).
// Strategy: fused stencil passes, LDS tiling fed by CDNA5 async global->LDS
// DMA (ASYNCcnt), closed-form boundary conditions instead of materialized
// padded arrays, multigrid ping-pong so final fields land in d_out, and an
// exact WMMA (V_WMMA_F32_16X16X4_F32) formulation of the stride-2 restriction
// (x-pair reduction as D = P(16x32) x F(32x16), P a 0/1 pairing matrix).
// Memory-bound on L2 (fields are 16MB each, global L2 = 192MB).
// ---------------------------------------------------------------------------

#define NZd 64
#define NYd 256
#define NXd 256
#define NTOT (NZd * NYd * NXd)
#define RE_C 0.15f
#define UB_C (-1.0f)
#define ITER_C 2          // reference ITERATION (device scalar unreadable during capture)
#define INV26 (1.0f / 26.0f)

#define USE_ASYNC_LDS 1   // CDNA5 async global->LDS DMA path (probe-verified round 1)

typedef float v2f __attribute__((ext_vector_type(2)));
typedef float v8f __attribute__((ext_vector_type(8)));

// ---- indexing -------------------------------------------------------------
__device__ __forceinline__ int gidx(int z, int y, int x) { return (z * NYd + y) * NXd + x; }

// ---- closed-form boundary-condition fetchers ------------------------------
// Derived from the sequential .at[] updates in the reference (order matters).
__device__ __forceinline__ float fetch_bcp(const float* p, int ez, int ey, int ex) {
  if (ex >= NXd) return 0.0f;                 // x-max Dirichlet 0
  ex = ex < 0 ? 0 : ex;                       // x-min Neumann
  ey = ey < 0 ? 0 : (ey >= NYd ? NYd - 1 : ey);
  ez = ez < 0 ? 0 : (ez >= NZd ? NZd - 1 : ez);
  return p[gidx(ez, ey, ex)];
}
__device__ __forceinline__ float fetch_bcu(const float* u, int ez, int ey, int ex) {
  if (ez < 0) return 0.0f;                    // z=0 plane zeroed last -> wins
  if (ex < 0 || ex >= NXd) return UB_C;       // inflow/outflow faces
  ez = ez >= NZd ? NZd - 1 : ez;
  ey = ey < 0 ? 0 : (ey >= NYd ? NYd - 1 : ey);
  return u[gidx(ez, ey, ex)];
}
__device__ __forceinline__ float fetch_bcv(const float* v, int ez, int ey, int ex) {
  if (ex < 0 || ex >= NXd || ey < 0 || ey >= NYd || ez < 0) return 0.0f;
  ez = ez >= NZd ? NZd - 1 : ez;
  return v[gidx(ez, ey, ex)];
}
__device__ __forceinline__ float fetch_bcw(const float* w, int ez, int ey, int ex) {
  if (ex < 0 || ex >= NXd || ey < 0 || ey >= NYd || ez < 0 || ez >= NZd) return 0.0f;
  return w[gidx(ez, ey, ex)];
}
// zero-pad fetch on arbitrary dims (multigrid coarse levels)
__device__ __forceinline__ float zp(const float* f, int z, int y, int x, int nz, int ny, int nx) {
  if ((unsigned)z >= (unsigned)nz || (unsigned)y >= (unsigned)ny || (unsigned)x >= (unsigned)nx)
    return 0.0f;
  return f[(z * ny + y) * nx + x];
}

// ---- CDNA5 async global->LDS helpers --------------------------------------
__device__ __forceinline__ void async_ld_b32(float* lds_dst, const float* gbase, unsigned byte_off) {
#if USE_ASYNC_LDS
  // GVS mode: mem = SGPR_base + VGPR_u32_offset.  VDST holds LDS byte address
  // (low 32 bits of the flat LDS pointer are the LDS offset per aperture rules).
  unsigned d = (unsigned)(uintptr_t)lds_dst;
  unsigned long long b = (unsigned long long)(uintptr_t)gbase;
  asm volatile("global_load_async_to_lds_b32 %0, %1, %2"
               :: "v"(d), "v"(byte_off), "s"(b) : "memory");
#else
  *lds_dst = *(const float*)((const char*)gbase + byte_off);
#endif
}
__device__ __forceinline__ void async_wait_all() {
#if USE_ASYNC_LDS
  asm volatile("s_wait_asynccnt 0" ::: "memory");
#endif
}

// ---- tile geometry for fused stencil kernels ------------------------------
#define TLZ 8
#define TLY 8
#define TLX 32
#define HLZ (TLZ + 2)
#define HLY (TLY + 2)
#define HLX (TLX + 2)
#define HTOT (HLZ * HLY * HLX)   // 3400 floats
__device__ __forceinline__ int lidx(int lz, int ly, int lx) { return (lz * HLY + ly) * HLX + lx; }

struct Ads { float adx, ady, adz, ad2, c; };

__device__ __forceinline__ Ads stencil27(const float* s, int lz, int ly, int lx) {
  float sum = 0.0f;
#pragma unroll
  for (int dz = -1; dz <= 1; ++dz)
#pragma unroll
    for (int dy = -1; dy <= 1; ++dy)
#pragma unroll
      for (int dx = -1; dx <= 1; ++dx)
        sum += s[lidx(lz + dz, ly + dy, lx + dx)];
  Ads a;
  a.c   = s[lidx(lz, ly, lx)];
  a.ad2 = (sum - a.c) * INV26 - a.c;                 // conv with wA (= w1, DX=1)
  a.adx = 0.5f * (s[lidx(lz, ly, lx + 1)] - s[lidx(lz, ly, lx - 1)]);
  a.ady = 0.5f * (s[lidx(lz, ly + 1, lx)] - s[lidx(lz, ly - 1, lx)]);
  a.adz = 0.5f * (s[lidx(lz + 1, ly, lx)] - s[lidx(lz - 1, ly, lx)]);
  return a;
}

__device__ __forceinline__ float kd_val(float speed, const Ads& f) {
  float num = 0.1f * fabsf((1.0f / 3.0f) * speed * f.ad2);
  float den = 0.001f + (fabsf(f.adx) + fabsf(f.ady) + fabsf(f.adz)) * (1.0f / 3.0f);
  return num / den;
}

// ===========================================================================
// K1: solid body + copy p + pressure gradient of original p
// ===========================================================================
__global__ void k_prep(const float* __restrict__ u0, const float* __restrict__ v0,
                       const float* __restrict__ w0, const float* __restrict__ p0,
                       const float* __restrict__ sigma, const float* __restrict__ dtp,
                       float* __restrict__ U, float* __restrict__ V, float* __restrict__ W,
                       float* __restrict__ P,
                       float* __restrict__ Gpx, float* __restrict__ Gpy, float* __restrict__ Gpz) {
  int i = blockIdx.x * blockDim.x + threadIdx.x;
  if (i >= NTOT) return;
  int x = i % NXd, y = (i / NXd) % NYd, z = i / (NXd * NYd);
  float dt = dtp[0];
  float s = 1.0f / (1.0f + dt * sigma[i]);
  U[i] = u0[i] * s;  V[i] = v0[i] * s;  W[i] = w0[i] * s;  P[i] = p0[i];
  Gpx[i] = 0.5f * dt * (fetch_bcp(p0, z, y, x + 1) - fetch_bcp(p0, z, y, x - 1));
  Gpy[i] = 0.5f * dt * (fetch_bcp(p0, z, y + 1, x) - fetch_bcp(p0, z, y - 1, x));
  Gpz[i] = 0.5f * dt * (fetch_bcp(p0, z + 1, y, x) - fetch_bcp(p0, z - 1, y, x));
}

// ===========================================================================
// K2: fused advection/diffusion/k pass (predictor hfac=0.5, corrector hfac=1)
//  tmp_f = base + hfac*dt*(Re*AD2 - adv) + 0.5*hfac*dt*k*AD2 - Gp
// LDS tiles staged by async DMA with clamped addresses; BC values patched in.
// ===========================================================================
__global__ __launch_bounds__(256) void k_ads(
    const float* __restrict__ au, const float* __restrict__ av, const float* __restrict__ aw,
    const float* __restrict__ baseU, const float* __restrict__ baseV, const float* __restrict__ baseW,
    const float* __restrict__ Gpx, const float* __restrict__ Gpy, const float* __restrict__ Gpz,
    const float* __restrict__ k1, const float* __restrict__ sigma,
    const float* __restrict__ dtp, float hfac,
    float* __restrict__ TMPu, float* __restrict__ TMPv, float* __restrict__ TMPw,
    float* __restrict__ Ku, float* __restrict__ Kv, float* __restrict__ Kw) {
  __shared__ float su[HTOT], sv[HTOT], sw[HTOT];
  const int tx = threadIdx.x, ty = threadIdx.y;
  const int tid = ty * TLX + tx;
  const int x0 = blockIdx.x * TLX, y0 = blockIdx.y * TLY, z0 = blockIdx.z * TLZ;

  // stage tiles: clamped-address async DMA (one address serves all 3 fields)
  for (int t = tid; t < HTOT; t += 256) {
    int lz = t / (HLY * HLX), r = t % (HLY * HLX), ly = r / HLX, lx = r % HLX;
    int ez = z0 - 1 + lz, ey = y0 - 1 + ly, ex = x0 - 1 + lx;
    int cz = ez < 0 ? 0 : (ez >= NZd ? NZd - 1 : ez);
    int cy = ey < 0 ? 0 : (ey >= NYd ? NYd - 1 : ey);
    int cx = ex < 0 ? 0 : (ex >= NXd ? NXd - 1 : ex);
    unsigned off = (unsigned)gidx(cz, cy, cx) * 4u;
    async_ld_b32(&su[t], au, off);
    async_ld_b32(&sv[t], av, off);
    async_ld_b32(&sw[t], aw, off);
  }
  async_wait_all();
  __syncthreads();
  // patch boundary values (the only non-clamp parts of bc_u / bc_v / bc_w)
  for (int t = tid; t < HTOT; t += 256) {
    int lz = t / (HLY * HLX), r = t % (HLY * HLX), ly = r / HLX, lx = r % HLX;
    int ez = z0 - 1 + lz, ey = y0 - 1 + ly, ex = x0 - 1 + lx;
    bool xh = (ex < 0) | (ex >= NXd);
    bool yh = (ey < 0) | (ey >= NYd);
    bool zlo = ez < 0, zhi = ez >= NZd;
    if (zlo) su[t] = 0.0f; else if (xh) su[t] = UB_C;
    if (xh | yh | zlo)       sv[t] = 0.0f;
    if (xh | yh | zlo | zhi) sw[t] = 0.0f;
  }
  __syncthreads();

  const float dt = dtp[0];
  const float hdt = hfac * dt;
  for (int zz = 0; zz < TLZ; ++zz) {
    int lz = zz + 1, ly = ty + 1, lx = tx + 1;
    int g = gidx(z0 + zz, y0 + ty, x0 + tx);
    Ads Au = stencil27(su, lz, ly, lx);
    Ads Av = stencil27(sv, lz, ly, lx);
    Ads Aw = stencil27(sw, lz, ly, lx);
    float sinv = 1.0f / (1.0f + dt * sigma[g]);
    float k1c = k1[g];
    float speed = fabsf(Au.c) + fabsf(Av.c) + fabsf(Aw.c);
    float ku = fminf(kd_val(speed, Au), k1c) * sinv;
    float kv = fminf(kd_val(speed, Av), k1c) * sinv;
    float kw = fminf(kd_val(speed, Aw), k1c) * sinv;
    float adv_u = Au.c * Au.adx + Av.c * Au.ady + Aw.c * Au.adz;
    float adv_v = Au.c * Av.adx + Av.c * Av.ady + Aw.c * Av.adz;
    float adv_w = Au.c * Aw.adx + Av.c * Aw.ady + Aw.c * Aw.adz;
    TMPu[g] = baseU[g] + hdt * (RE_C * Au.ad2 - adv_u) + 0.5f * hdt * ku * Au.ad2 - Gpx[g];
    TMPv[g] = baseV[g] + hdt * (RE_C * Av.ad2 - adv_v) + 0.5f * hdt * kv * Av.ad2 - Gpy[g];
    TMPw[g] = baseW[g] + hdt * (RE_C * Aw.ad2 - adv_w) + 0.5f * hdt * kw * Aw.ad2 - Gpz[g];
    Ku[g] = ku;  Kv[g] = kv;  Kw[g] = kw;
  }
}

// ===========================================================================
// K3: nonlocal k term + solid body:
//  out = (tmp + 0.5*hfac*dt*(conv0(a*k) - a_c*conv0(k))) / (1+dt*sigma)
// conv0 = zero-padded wA conv (boundary_condition_k zeroes halos).
// ===========================================================================
__global__ __launch_bounds__(256) void k_finish(
    const float* __restrict__ af, const float* __restrict__ kf,
    const float* __restrict__ tmpf, const float* __restrict__ sigma,
    const float* __restrict__ dtp, float hfac, float* __restrict__ outf) {
  __shared__ float sa[HTOT], sk[HTOT];
  const int tx = threadIdx.x, ty = threadIdx.y;
  const int tid = ty * TLX + tx;
  const int x0 = blockIdx.x * TLX, y0 = blockIdx.y * TLY, z0 = blockIdx.z * TLZ;

  for (int t = tid; t < HTOT; t += 256) {
    int lz = t / (HLY * HLX), r = t % (HLY * HLX), ly = r / HLX, lx = r % HLX;
    int ez = z0 - 1 + lz, ey = y0 - 1 + ly, ex = x0 - 1 + lx;
    int cz = ez < 0 ? 0 : (ez >= NZd ? NZd - 1 : ez);
    int cy = ey < 0 ? 0 : (ey >= NYd ? NYd - 1 : ey);
    int cx = ex < 0 ? 0 : (ex >= NXd ? NXd - 1 : ex);
    unsigned off = (unsigned)gidx(cz, cy, cx) * 4u;
    async_ld_b32(&sa[t], af, off);
    async_ld_b32(&sk[t], kf, off);
  }
  async_wait_all();
  __syncthreads();
  for (int t = tid; t < HTOT; t += 256) {
    int lz = t / (HLY * HLX), r = t % (HLY * HLX), ly = r / HLX, lx = r % HLX;
    int ez = z0 - 1 + lz, ey = y0 - 1 + ly, ex = x0 - 1 + lx;
    if ((ex < 0) | (ex >= NXd) | (ey < 0) | (ey >= NYd) | (ez < 0) | (ez >= NZd)) {
      sa[t] = 0.0f; sk[t] = 0.0f;   // k zero-padded; a*k halo product is 0 anyway
    }
  }
  __syncthreads();

  const float dt = dtp[0];
  const float hdt = hfac * dt;
  for (int zz = 0; zz < TLZ; ++zz) {
    int lz = zz + 1, ly = ty + 1, lx = tx + 1;
    int g = gidx(z0 + zz, y0 + ty, x0 + tx);
    float sumA = 0.0f, sumK = 0.0f;
#pragma unroll
    for (int dz = -1; dz <= 1; ++dz)
#pragma unroll
      for (int dy = -1; dy <= 1; ++dy)
#pragma unroll
        for (int dx = -1; dx <= 1; ++dx) {
          float av = sa[lidx(lz + dz, ly + dy, lx + dx)];
          float kv = sk[lidx(lz + dz, ly + dy, lx + dx)];
          sumA += av * kv;  sumK += kv;
        }
    float ac = sa[lidx(lz, ly, lx)], kc = sk[lidx(lz, ly, lx)];
    float pc = ac * kc;
    float convA = (sumA - pc) * INV26 - pc;
    float convK = (sumK - kc) * INV26 - kc;
    float sinv = 1.0f / (1.0f + dt * sigma[g]);
    outf[g] = (tmpf[g] + 0.5f * hdt * (convA - ac * convK)) * sinv;
  }
}

// ===========================================================================
// Multigrid helpers
// ===========================================================================
__global__ void k_div(const float* __restrict__ U, const float* __restrict__ V,
                      const float* __restrict__ W, const float* __restrict__ dtp,
                      float* __restrict__ bb) {
  int i = blockIdx.x * blockDim.x + threadIdx.x;
  if (i >= NTOT) return;
  int x = i % NXd, y = (i / NXd) % NYd, z = i / (NXd * NYd);
  float div = 0.5f * (fetch_bcu(U, z, y, x + 1) - fetch_bcu(U, z, y, x - 1))
            + 0.5f * (fetch_bcv(V, z, y + 1, x) - fetch_bcv(V, z, y - 1, x))
            + 0.5f * (fetch_bcw(W, z + 1, y, x) - fetch_bcw(W, z - 1, y, x));
  bb[i] = -div / dtp[0];
}

__global__ void k_lapres(const float* __restrict__ p, const float* __restrict__ bb,
                         float* __restrict__ r) {
  int i = blockIdx.x * blockDim.x + threadIdx.x;
  if (i >= NTOT) return;
  int x = i % NXd, y = (i / NXd) % NYd, z = i / (NXd * NYd);
  if (z + 1 < NZd) __builtin_prefetch(&p[i + NXd * NYd], 0, 1);   // global_prefetch_b8
  float sum = 0.0f;
#pragma unroll
  for (int dz = -1; dz <= 1; ++dz)
#pragma unroll
    for (int dy = -1; dy <= 1; ++dy)
#pragma unroll
      for (int dx = -1; dx <= 1; ++dx)
        sum += fetch_bcp(p, z + dz, y + dy, x + dx);
  float c = p[i];
  r[i] = (sum - c) * INV26 - c - bb[i];
}

__global__ void k_psmooth(const float* __restrict__ pin, const float* __restrict__ bb,
                          const float* __restrict__ wAp, float* __restrict__ pout) {
  int i = blockIdx.x * blockDim.x + threadIdx.x;
  if (i >= NTOT) return;
  int x = i % NXd, y = (i / NXd) % NYd, z = i / (NXd * NYd);
  float sum = 0.0f;
#pragma unroll
  for (int dz = -1; dz <= 1; ++dz)
#pragma unroll
    for (int dy = -1; dy <= 1; ++dy)
#pragma unroll
      for (int dx = -1; dx <= 1; ++dx)
        sum += fetch_bcp(pin, z + dz, y + dy, x + dx);
  float c = pin[i];
  float lap = (sum - c) * INV26 - c;
  float invd = 1.0f / wAp[13];            // wA[1][1][1]
  pout[i] = c - lap * invd + bb[i] * invd;
}

// VALU restriction (used for levels whose dims don't fit the WMMA tile)
__global__ void k_restrict(const float* __restrict__ src, float* __restrict__ dst,
                           int cnz, int cny, int cnx) {
  int i = blockIdx.x * blockDim.x + threadIdx.x;
  int n = cnz * cny * cnx;
  if (i >= n) return;
  int x = i % cnx, y = (i / cnx) % cny, z = i / (cnx * cny);
  int fny = cny * 2, fnx = cnx * 2;
  int b0 = ((2 * z) * fny + 2 * y) * fnx + 2 * x;
  int b1 = b0 + fny * fnx;
  float sum = src[b0] + src[b0 + 1] + src[b0 + fnx] + src[b0 + fnx + 1]
            + src[b1] + src[b1 + 1] + src[b1 + fnx] + src[b1 + fnx + 1];
  dst[i] = 0.125f * sum;
}

// ===========================================================================
// WMMA restriction: one wave computes a 16(x) x 8(y) x 1(z) coarse tile.
// The x-direction pair reduction is exact 0/1-matrix math:
//   D(16x16) = P(16x32) x F(32x16),  P[m][k] = (k>>1 == m)
// run as 8 chained V_WMMA_F32_16X16X4_F32 (K accumulated through C).
// Operand layouts per CDNA5 ISA 7.12.2 (wave32):
//   A 16x4:  lane M=L%16; VGPR0/1 = K{0,1} (L<16) or K{2,3} (L>=16)
//   B 4x16:  lane N=L%16; VGPR0/1 = K{0,1} (L<16) or K{2,3} (L>=16)
//   C/D:     lane N=L%16; VGPR i -> M = i + 8*(L>=16)
// y/z pair combine done on the LDS-staged D.  Same sums as the VALU kernel
// (x-pairs added first, so rounding can differ; same 0.125 scale).
// ===========================================================================
__global__ __launch_bounds__(32) void k_restrict_wmma(
    const float* __restrict__ src, float* __restrict__ dst,
    int cnz, int cny, int cnx) {
  __shared__ float sD[2][16 * 16];
  const int lane = threadIdx.x;            // 0..31, one full wave (EXEC all 1s)
  const int xc0 = blockIdx.x * 16;
  const int yc0 = blockIdx.y * 8;
  const int zc  = blockIdx.z;
  const int fny = cny * 2, fnx = cnx * 2;
  const int xf0 = xc0 * 2, yf0 = yc0 * 2;
  const int n    = lane & 15;
  const int half = lane >> 4;

#if defined(__HIP_DEVICE_COMPILE__) && __has_builtin(__builtin_amdgcn_wmma_f32_16x16x4_f32)
  const int m = n;                          // A-matrix M index for this lane
  for (int s = 0; s < 2; ++s) {             // fine z plane pair
    const float* rowp = src + ((size_t)(zc * 2 + s) * fny + (yf0 + n)) * fnx + xf0;
    v8f c = {0.0f, 0.0f, 0.0f, 0.0f, 0.0f, 0.0f, 0.0f, 0.0f};
#pragma unroll
    for (int kb = 0; kb < 8; ++kb) {
      int kbase = kb * 4 + half * 2;        // this lane's global fine-x K pair
      v2f a, b;
      a.x = ((kbase >> 1) == m) ? 1.0f : 0.0f;        // P[m][kbase]
      a.y = (((kbase + 1) >> 1) == m) ? 1.0f : 0.0f;  // P[m][kbase+1]
      b.x = rowp[kbase];                               // F[kbase][n]
      b.y = rowp[kbase + 1];                           // F[kbase+1][n]
      c = __builtin_amdgcn_wmma_f32_16x16x4_f32(false, a, false, b,
                                                (short)0, c, false, false);
    }
#pragma unroll
    for (int i = 0; i < 8; ++i)
      sD[s][(i + 8 * half) * 16 + n] = c[i];          // D(m = i+8*half, n)
  }
#else
  // host-pass / no-builtin fallback: same math, plain VALU
  for (int s = 0; s < 2; ++s) {
    const float* rowp = src + ((size_t)(zc * 2 + s) * fny) * fnx;
    for (int t = lane; t < 256; t += 32) {
      int mm = t >> 4, nn = t & 15;
      const float* rp = rowp + (size_t)(yf0 + nn) * fnx + xf0;
      sD[s][mm * 16 + nn] = rp[2 * mm] + rp[2 * mm + 1];
    }
  }
#endif
  __syncthreads();
  for (int t = lane; t < 128; t += 32) {
    int mm = t & 15, j = t >> 4;            // coarse x, coarse y within tile
    float v = sD[0][mm * 16 + 2 * j] + sD[0][mm * 16 + 2 * j + 1]
            + sD[1][mm * 16 + 2 * j] + sD[1][mm * 16 + 2 * j + 1];
    dst[((size_t)zc * cny + (yc0 + j)) * cnx + (xc0 + mm)] = 0.125f * v;
  }
}

__global__ void k_wsmooth(const float* __restrict__ win, const float* __restrict__ rr,
                          const float* __restrict__ wAp, float* __restrict__ wout,
                          int nz, int ny, int nx) {
  int i = blockIdx.x * blockDim.x + threadIdx.x;
  int n = nz * ny * nx;
  if (i >= n) return;
  int x = i % nx, y = (i / nx) % ny, z = i / (nx * ny);
  float sum = 0.0f;
#pragma unroll
  for (int dz = -1; dz <= 1; ++dz)
#pragma unroll
    for (int dy = -1; dy <= 1; ++dy)
#pragma unroll
      for (int dx = -1; dx <= 1; ++dx)
        sum += zp(win, z + dz, y + dy, x + dx, nz, ny, nx);
  float c = win[i];
  float lap = (sum - c) * INV26 - c;
  float invd = 1.0f / wAp[13];
  wout[i] = c - lap * invd + rr[i] * invd;
}

__global__ void k_prol(const float* __restrict__ cin, float* __restrict__ fout,
                       int fnz, int fny, int fnx) {
  int i = blockIdx.x * blockDim.x + threadIdx.x;
  int n = fnz * fny * fnx;
  if (i >= n) return;
  int x = i % fnx, y = (i / fnx) % fny, z = i / (fnx * fny);
  int cny = fny >> 1, cnx = fnx >> 1;
  fout[i] = cin[((z >> 1) * cny + (y >> 1)) * cnx + (x >> 1)];
}

__global__ void k_sub(float* __restrict__ p, const float* __restrict__ w, int n) {
  int i = blockIdx.x * blockDim.x + threadIdx.x;
  if (i < n) p[i] -= w[i];
}
__global__ void k_zero(float* __restrict__ p, int n) {
  int i = blockIdx.x * blockDim.x + threadIdx.x;
  if (i < n) p[i] = 0.0f;
}
__global__ void k_copy(float* __restrict__ d, const float* __restrict__ s, int n) {
  int i = blockIdx.x * blockDim.x + threadIdx.x;
  if (i < n) d[i] = s[i];
}

// final pressure-gradient correction + solid body (in place, center-only reads)
__global__ void k_pgc(const float* __restrict__ P, const float* __restrict__ sigma,
                      const float* __restrict__ dtp,
                      float* __restrict__ U, float* __restrict__ V, float* __restrict__ W) {
  int i = blockIdx.x * blockDim.x + threadIdx.x;
  if (i >= NTOT) return;
  int x = i % NXd, y = (i / NXd) % NYd, z = i / (NXd * NYd);
  float dt = dtp[0];
  float s = 1.0f / (1.0f + dt * sigma[i]);
  U[i] = (U[i] - 0.5f * dt * (fetch_bcp(P, z, y, x + 1) - fetch_bcp(P, z, y, x - 1))) * s;
  V[i] = (V[i] - 0.5f * dt * (fetch_bcp(P, z, y + 1, x) - fetch_bcp(P, z, y - 1, x))) * s;
  W[i] = (W[i] - 0.5f * dt * (fetch_bcp(P, z + 1, y, x) - fetch_bcp(P, z - 1, y, x))) * s;
}

// ===========================================================================
// Host driver
// ===========================================================================
extern "C" void kernel_launch(void* const* d_in, const int* in_sizes, int n_in,
                              void* d_out, int out_size, void* d_ws, size_t ws_size,
                              hipStream_t stream) {
  (void)in_sizes; (void)n_in; (void)out_size; (void)ws_size;
  const float* u0    = (const float*)d_in[0];
  const float* v0    = (const float*)d_in[2];
  const float* w0    = (const float*)d_in[4];
  const float* p0    = (const float*)d_in[6];
  const float* k1    = (const float*)d_in[11];
  const float* sigma = (const float*)d_in[15];
  const float* wAp   = (const float*)d_in[16];
  const float* dtp   = (const float*)d_in[22];

  const int N = NTOT;
  float* out = (float*)d_out;
  float* U = out;            float* V = out + (size_t)N;
  float* W = out + 2ull * N; float* P = out + 3ull * N;
  float* WMG  = out + 4ull * N;
  float* ROUT = out + 5ull * N;

  // workspace layout (floats); requires ws_size >= ~259 MB
  float* ws = (float*)d_ws;
  float* GPX = ws;               float* GPY = ws + 1ull * N;  float* GPZ = ws + 2ull * N;
  float* TMPU = ws + 3ull * N;   float* TMPV = ws + 4ull * N; float* TMPW = ws + 5ull * N;
  float* KU = ws + 6ull * N;     float* KV = ws + 7ull * N;   float* KW = ws + 8ull * N;
  float* BU = ws + 9ull * N;     float* BV = ws + 10ull * N;  float* BW = ws + 11ull * N;
  float* MGB = ws + 12ull * N;
  float* R0  = ws + 13ull * N;
  float* PT  = ws + 14ull * N;
  size_t o = 15ull * N;
  const int n1 = 32 * 128 * 128, n2 = 16 * 64 * 64, n3 = 8 * 32 * 32, n4 = 4 * 16 * 16, n5 = 2 * 8 * 8;
  float* R1 = ws + o; o += n1;   float* R2 = ws + o; o += n2;
  float* R3 = ws + o; o += n3;   float* R4 = ws + o; o += n4;
  float* R5 = ws + o; o += n5;
  float* WTA = ws + o; o += n1;  float* WTB = ws + o; o += n1;

  const int eB = (N + 255) / 256;
  dim3 tB(TLX, TLY, 1);
  dim3 tG(NXd / TLX, NYd / TLY, NZd / TLZ);
  auto nb = [](int n) { return (n + 255) / 256; };
  auto restrict_any = [&](const float* s, float* d, int cz, int cy, int cx) {
    if ((cx % 16 == 0) && (cy % 8 == 0)) {
      dim3 g(cx / 16, cy / 8, cz);
      k_restrict_wmma<<<g, 32, 0, stream>>>(s, d, cz, cy, cx);
    } else {
      k_restrict<<<(cz * cy * cx + 255) / 256, 256, 0, stream>>>(s, d, cz, cy, cx);
    }
  };

  // ---- prep: solid body, p copy, Gp ----
  k_prep<<<eB, 256, 0, stream>>>(u0, v0, w0, p0, sigma, dtp, U, V, W, P, GPX, GPY, GPZ);

  // ---- predictor half step ----
  k_ads<<<tG, tB, 0, stream>>>(U, V, W, U, V, W, GPX, GPY, GPZ, k1, sigma, dtp, 0.5f,
                               TMPU, TMPV, TMPW, KU, KV, KW);
  k_finish<<<tG, tB, 0, stream>>>(U, KU, TMPU, sigma, dtp, 0.5f, BU);
  k_finish<<<tG, tB, 0, stream>>>(V, KV, TMPV, sigma, dtp, 0.5f, BV);
  k_finish<<<tG, tB, 0, stream>>>(W, KW, TMPW, sigma, dtp, 0.5f, BW);

  // ---- corrector full step (advect with b fields, base = U,V,W) ----
  k_ads<<<tG, tB, 0, stream>>>(BU, BV, BW, U, V, W, GPX, GPY, GPZ, k1, sigma, dtp, 1.0f,
                               TMPU, TMPV, TMPW, KU, KV, KW);
  k_finish<<<tG, tB, 0, stream>>>(BU, KU, TMPU, sigma, dtp, 1.0f, U);
  k_finish<<<tG, tB, 0, stream>>>(BV, KV, TMPV, sigma, dtp, 1.0f, V);
  k_finish<<<tG, tB, 0, stream>>>(BW, KW, TMPW, sigma, dtp, 1.0f, W);

  // ---- multigrid F-cycle pressure solve ----
  k_div<<<eB, 256, 0, stream>>>(U, V, W, dtp, MGB);
  for (int it = 0; it < ITER_C; ++it) {
    float* pin  = (it == 0) ? P : PT;
    float* pout = (it == 0) ? PT : P;
    k_lapres<<<eB, 256, 0, stream>>>(pin, MGB, R0);
    restrict_any(R0, R1, 32, 128, 128);
    restrict_any(R1, R2, 16, 64, 64);
    restrict_any(R2, R3, 8, 32, 32);
    restrict_any(R3, R4, 4, 16, 16);
    restrict_any(R4, R5, 2, 8, 8);          // 8x8 plane -> VALU kernel
    // up-sweep: w starts at zero on the coarsest level
    k_zero<<<1, 256, 0, stream>>>(WTA, n5);
    k_wsmooth<<<nb(n5), 256, 0, stream>>>(WTA, R5, wAp, WTB, 2, 8, 8);
    k_prol<<<nb(n4), 256, 0, stream>>>(WTB, WTA, 4, 16, 16);
    k_wsmooth<<<nb(n4), 256, 0, stream>>>(WTA, R4, wAp, WTB, 4, 16, 16);
    k_prol<<<nb(n3), 256, 0, stream>>>(WTB, WTA, 8, 32, 32);
    k_wsmooth<<<nb(n3), 256, 0, stream>>>(WTA, R3, wAp, WTB, 8, 32, 32);
    k_prol<<<nb(n2), 256, 0, stream>>>(WTB, WTA, 16, 64, 64);
    k_wsmooth<<<nb(n2), 256, 0, stream>>>(WTA, R2, wAp, WTB, 16, 64, 64);
    k_prol<<<nb(n1), 256, 0, stream>>>(WTB, WTA, 32, 128, 128);
    k_wsmooth<<<nb(n1), 256, 0, stream>>>(WTA, R1, wAp, WTB, 32, 128, 128);
    k_prol<<<eB, 256, 0, stream>>>(WTB, WMG, 64, 256, 256);
    k_sub<<<eB, 256, 0, stream>>>(pin, WMG, N);
    k_psmooth<<<eB, 256, 0, stream>>>(pin, MGB, wAp, pout);
  }
  k_copy<<<1, 256, 0, stream>>>(ROUT, R5, n5);

  // ---- pressure-gradient velocity correction + solid body ----
  k_pgc<<<eB, 256, 0, stream>>>(P, sigma, dtp, U, V, W);
}